// MambaBlock_81810537054457
// MI455X (gfx1250) — compile-verified
//
#include <hip/hip_runtime.h>
#include <hip/hip_bf16.h>
#include <math.h>

#define BATCH  2
#define SEQ    2048
#define DMODEL 1024
#define DFF    4096
#define DSTATE 8
#define DTRANK 64
#define BL     (BATCH * SEQ)
#define PROJW  (DTRANK + 2 * DSTATE)   // 80
#define NCHUNK 32
#define CLEN   (SEQ / NCHUNK)          // 64
#define BD     (BATCH * DMODEL)        // 2048

typedef __attribute__((ext_vector_type(2))) float v2f;
typedef __attribute__((ext_vector_type(8))) float v8f;

#define WMMA4(A_, B_, C_) \
    __builtin_amdgcn_wmma_f32_16x16x4_f32(false, (A_), false, (B_), (short)0, (C_), false, false)

// ---------------------------------------------------------------------------
// Transpose (B, D, L) -> (B, L, D) via LDS tile
// ---------------------------------------------------------------------------
__global__ void transpose_bdl_to_bld(const float* __restrict__ x, float* __restrict__ xt) {
    __shared__ float tile[32][33];
    const int b  = blockIdx.z;
    const int l0 = blockIdx.x * 32;
    const int d0 = blockIdx.y * 32;
    const int tx = threadIdx.x, ty = threadIdx.y;  // 32 x 8
    const size_t xbase = (size_t)b * DMODEL * SEQ;
    #pragma unroll
    for (int j = 0; j < 32; j += 8)
        tile[ty + j][tx] = x[xbase + (size_t)(d0 + ty + j) * SEQ + (l0 + tx)];
    __syncthreads();
    const size_t obase = (size_t)b * SEQ * DMODEL;
    #pragma unroll
    for (int j = 0; j < 32; j += 8)
        xt[obase + (size_t)(l0 + ty + j) * DMODEL + (d0 + tx)] = tile[tx][ty + j];
}

// ---------------------------------------------------------------------------
// fp32 WMMA GEMM: C[M,Nout] = A[M,K] @ W[Nout,K]^T (+bias, +relu)
// One wave computes a 32x64 block (2 m-tiles x 4 n-tiles) with K-step 4 via
// V_WMMA_F32_16X16X4_F32. Branch-free body: tail tiles clamp their column to
// Nout-16 and recompute identical values (benign same-value stores), so EXEC
// stays all-1s across every WMMA. Software-pipelined: fragments for k+4 are
// loaded while the 8 WMMAs for k issue.
// A/B frag: lane l -> row = l&15, k = 2*(l>>4)+v ; D: n = l&15, m = r+8*(l>>4)
// ---------------------------------------------------------------------------
__device__ __forceinline__ void store_tile(v8f acc, float* __restrict__ C, int ldc,
                                           int mb, int n0, const float* __restrict__ bias,
                                           int act) {
    const float bv = bias ? bias[n0] : 0.f;
    float* cp = C + (size_t)mb * ldc + n0;
    #pragma unroll
    for (int r = 0; r < 8; ++r) {
        float v = acc[r] + bv;
        if (act) v = fmaxf(v, 0.f);
        cp[(size_t)r * ldc] = v;
    }
}

__global__ __launch_bounds__(128) void gemm_wmma_f32(
    const float* __restrict__ A, int lda,
    const float* __restrict__ W, int ldw,
    float* __restrict__ C, int ldc,
    const float* __restrict__ bias,
    int K, int Nout, int act) {
    const int lane = threadIdx.x & 31;
    const int wave = threadIdx.x >> 5;
    const int lm   = lane & 15;
    const int kh   = (lane >> 4) << 1;          // 0 or 2
    const int m0   = blockIdx.y * 32;
    const int ncb  = blockIdx.x * 256 + wave * 64;
    const int nmax = Nout - 16;
    int c0 = ncb;      c0 = (c0 < nmax) ? c0 : nmax;
    int c1 = ncb + 16; c1 = (c1 < nmax) ? c1 : nmax;
    int c2 = ncb + 32; c2 = (c2 < nmax) ? c2 : nmax;
    int c3 = ncb + 48; c3 = (c3 < nmax) ? c3 : nmax;

    const float* ap0 = A + (size_t)(m0 + lm) * lda + kh;
    const float* ap1 = ap0 + (size_t)16 * lda;
    const float* w0  = W + (size_t)(c0 + lm) * ldw + kh;
    const float* w1  = W + (size_t)(c1 + lm) * ldw + kh;
    const float* w2  = W + (size_t)(c2 + lm) * ldw + kh;
    const float* w3  = W + (size_t)(c3 + lm) * ldw + kh;

    v8f d00 = {}, d01 = {}, d02 = {}, d03 = {};
    v8f d10 = {}, d11 = {}, d12 = {}, d13 = {};

    v2f a0 = *(const v2f*)ap0;
    v2f a1 = *(const v2f*)ap1;
    v2f b0 = *(const v2f*)w0;
    v2f b1 = *(const v2f*)w1;
    v2f b2 = *(const v2f*)w2;
    v2f b3 = *(const v2f*)w3;

    for (int k0 = 4; k0 < K; k0 += 4) {
        if (((k0 & 63) == 4) && (k0 + 64 < K)) {   // -> global_prefetch_b8
            __builtin_prefetch(ap0 + k0 + 64, 0, 0);
            __builtin_prefetch(ap1 + k0 + 64, 0, 0);
            __builtin_prefetch(w0  + k0 + 64, 0, 0);
            __builtin_prefetch(w3  + k0 + 64, 0, 0);
        }
        v2f a0n = *(const v2f*)(ap0 + k0);
        v2f a1n = *(const v2f*)(ap1 + k0);
        v2f b0n = *(const v2f*)(w0 + k0);
        v2f b1n = *(const v2f*)(w1 + k0);
        v2f b2n = *(const v2f*)(w2 + k0);
        v2f b3n = *(const v2f*)(w3 + k0);
        d00 = WMMA4(a0, b0, d00);
        d01 = WMMA4(a0, b1, d01);
        d02 = WMMA4(a0, b2, d02);
        d03 = WMMA4(a0, b3, d03);
        d10 = WMMA4(a1, b0, d10);
        d11 = WMMA4(a1, b1, d11);
        d12 = WMMA4(a1, b2, d12);
        d13 = WMMA4(a1, b3, d13);
        a0 = a0n; a1 = a1n;
        b0 = b0n; b1 = b1n; b2 = b2n; b3 = b3n;
    }
    d00 = WMMA4(a0, b0, d00);
    d01 = WMMA4(a0, b1, d01);
    d02 = WMMA4(a0, b2, d02);
    d03 = WMMA4(a0, b3, d03);
    d10 = WMMA4(a1, b0, d10);
    d11 = WMMA4(a1, b1, d11);
    d12 = WMMA4(a1, b2, d12);
    d13 = WMMA4(a1, b3, d13);

    const int mb0 = m0 + ((lane >> 4) << 3);
    const int mb1 = mb0 + 16;
    store_tile(d00, C, ldc, mb0, c0 + lm, bias, act);
    store_tile(d01, C, ldc, mb0, c1 + lm, bias, act);
    store_tile(d02, C, ldc, mb0, c2 + lm, bias, act);
    store_tile(d03, C, ldc, mb0, c3 + lm, bias, act);
    store_tile(d10, C, ldc, mb1, c0 + lm, bias, act);
    store_tile(d11, C, ldc, mb1, c1 + lm, bias, act);
    store_tile(d12, C, ldc, mb1, c2 + lm, bias, act);
    store_tile(d13, C, ldc, mb1, c3 + lm, bias, act);
}

// ---------------------------------------------------------------------------
// Depthwise causal conv (k=2) + SiLU. dir=0: prev = l-1; dir=1 (reversed
// sequence run in-place): prev = l+1. Out-of-range neighbor = 0.
// ---------------------------------------------------------------------------
__global__ void conv_silu_kernel(const float* __restrict__ xz, const float* __restrict__ cw,
                                 const float* __restrict__ cb, float* __restrict__ xc, int dir) {
    const int i  = blockIdx.x * blockDim.x + threadIdx.x;  // over BL*DMODEL
    const int d  = i % DMODEL;
    const int bl = i / DMODEL;
    const int l  = bl % SEQ;
    const int lp = dir ? (l + 1) : (l - 1);
    const float xi = xz[(size_t)bl * (2 * DMODEL) + d];
    float xp = 0.f;
    if (lp >= 0 && lp < SEQ)
        xp = xz[(size_t)(bl - l + lp) * (2 * DMODEL) + d];
    const float v = xp * cw[d * 2 + 0] + xi * cw[d * 2 + 1] + cb[d];
    xc[i] = v / (1.f + expf(-v));
}

__global__ void softplus_kernel(float* p, int n) {
    const int i = blockIdx.x * blockDim.x + threadIdx.x;
    if (i < n) {
        const float x = p[i];
        p[i] = (x > 20.f) ? x : log1pf(expf(x));
    }
}

// ---------------------------------------------------------------------------
// Chunked selective scan, 3 phases. h_t = dA_t * h_{t-1} + dBx_t is affine in
// h, so each chunk of CLEN steps composes to (aP, bC): h_out = aP*h_in + bC.
// Phase1: per-(b,d,chunk) compute (aP, bC)          [BD*NCHUNK threads]
// Phase2: per-(b,d) scan over chunks -> h0 per chunk [BD threads, tiny]
// Phase3: per-(b,d,chunk) replay with known h0 -> y  [BD*NCHUNK threads]
// Thread layout keeps d innermost (bd = tid % BD) for coalescing.
// ---------------------------------------------------------------------------
__global__ void scan_phase1(const float* __restrict__ delta, const float* __restrict__ xc,
                            const float* __restrict__ proj, const float* __restrict__ A_log,
                            float* __restrict__ aP, float* __restrict__ bC, int dir) {
    const int tid = blockIdx.x * blockDim.x + threadIdx.x;  // BD*NCHUNK
    const int bd = tid % BD;
    const int c  = tid / BD;
    const int b = bd / DMODEL, d = bd % DMODEL;
    float Ac[DSTATE], a[DSTATE], bb[DSTATE];
    #pragma unroll
    for (int n = 0; n < DSTATE; ++n) {
        Ac[n] = -expf(A_log[(size_t)d * DSTATE + n]);
        a[n] = 1.f; bb[n] = 0.f;
    }
    for (int tt = 0; tt < CLEN; ++tt) {
        const int t = c * CLEN + tt;
        const int l = dir ? (SEQ - 1 - t) : t;
        const size_t row = (size_t)b * SEQ + l;
        const float dl = delta[row * DMODEL + d];
        const float xv = xc[row * DMODEL + d];
        const float* pr = proj + row * PROJW;
        #pragma unroll
        for (int n = 0; n < DSTATE; ++n) {
            const float dA  = expf(dl * Ac[n]);
            const float dBx = dl * pr[DTRANK + n] * xv;
            a[n]  = a[n] * dA;
            bb[n] = dA * bb[n] + dBx;
        }
    }
    const size_t o = (size_t)tid * DSTATE;
    #pragma unroll
    for (int n = 0; n < DSTATE; ++n) { aP[o + n] = a[n]; bC[o + n] = bb[n]; }
}

__global__ void scan_phase2(const float* __restrict__ aP, const float* __restrict__ bC,
                            float* __restrict__ h0) {
    const int bd = blockIdx.x * blockDim.x + threadIdx.x;  // BD
    float h[DSTATE];
    #pragma unroll
    for (int n = 0; n < DSTATE; ++n) h[n] = 0.f;
    for (int c = 0; c < NCHUNK; ++c) {
        const size_t o = ((size_t)c * BD + bd) * DSTATE;
        #pragma unroll
        for (int n = 0; n < DSTATE; ++n) {
            h0[o + n] = h[n];
            h[n] = aP[o + n] * h[n] + bC[o + n];
        }
    }
}

__global__ void scan_phase3(const float* delta, const float* __restrict__ xc,
                            const float* __restrict__ proj, const float* __restrict__ xz,
                            const float* __restrict__ A_log, const float* __restrict__ Dskip,
                            const float* __restrict__ h0, float* y, int dir) {
    const int tid = blockIdx.x * blockDim.x + threadIdx.x;  // BD*NCHUNK
    const int bd = tid % BD;
    const int c  = tid / BD;
    const int b = bd / DMODEL, d = bd % DMODEL;
    float Ac[DSTATE], h[DSTATE];
    const size_t o = (size_t)tid * DSTATE;
    #pragma unroll
    for (int n = 0; n < DSTATE; ++n) {
        Ac[n] = -expf(A_log[(size_t)d * DSTATE + n]);
        h[n]  = h0[o + n];
    }
    const float Dk = Dskip[d];
    for (int tt = 0; tt < CLEN; ++tt) {
        const int t = c * CLEN + tt;
        const int l = dir ? (SEQ - 1 - t) : t;
        const size_t row = (size_t)b * SEQ + l;
        const float dl = delta[row * DMODEL + d];
        const float xv = xc[row * DMODEL + d];
        const float* pr = proj + row * PROJW;
        float acc = 0.f;
        #pragma unroll
        for (int n = 0; n < DSTATE; ++n) {
            const float dA = expf(dl * Ac[n]);
            h[n] = dA * h[n] + dl * pr[DTRANK + n] * xv;
            acc  = fmaf(h[n], pr[DTRANK + DSTATE + n], acc);
        }
        const float zv = xz[row * (2 * DMODEL) + DMODEL + d];
        const float sz = zv / (1.f + expf(-zv));
        y[row * DMODEL + d] = (acc + Dk * xv) * sz;  // y aliases delta: same-thread RAW only
    }
}

// ---------------------------------------------------------------------------
// out[row] = LN(a[row] + r[row]) * g + beta   (accumulate=1: out += ...)
// ---------------------------------------------------------------------------
__global__ void ln_residual_kernel(const float* __restrict__ a, const float* __restrict__ r,
                                   const float* __restrict__ g, const float* __restrict__ be,
                                   float* out, int accumulate) {
    __shared__ float red[256];
    const int tid = threadIdx.x;
    const size_t base = (size_t)blockIdx.x * DMODEL;
    float v[4];
    float s = 0.f;
    #pragma unroll
    for (int i = 0; i < 4; ++i) {
        const int d = tid + i * 256;
        v[i] = a[base + d] + r[base + d];
        s += v[i];
    }
    red[tid] = s;
    __syncthreads();
    for (int off = 128; off > 0; off >>= 1) {
        if (tid < off) red[tid] += red[tid + off];
        __syncthreads();
    }
    const float mu = red[0] * (1.f / DMODEL);
    __syncthreads();
    float s2 = 0.f;
    #pragma unroll
    for (int i = 0; i < 4; ++i) { const float t = v[i] - mu; s2 += t * t; }
    red[tid] = s2;
    __syncthreads();
    for (int off = 128; off > 0; off >>= 1) {
        if (tid < off) red[tid] += red[tid + off];
        __syncthreads();
    }
    const float rstd = rsqrtf(red[0] * (1.f / DMODEL) + 1e-5f);
    #pragma unroll
    for (int i = 0; i < 4; ++i) {
        const int d = tid + i * 256;
        const float ov = (v[i] - mu) * rstd * g[d] + be[d];
        if (accumulate) out[base + d] += ov;
        else            out[base + d]  = ov;
    }
}

// ---------------------------------------------------------------------------
static void launch_gemm(const float* A, int lda, const float* W, int ldw,
                        float* C, int ldc, const float* bias,
                        int M, int K, int Nout, int act, hipStream_t s) {
    dim3 grid((Nout + 255) / 256, M / 32);
    gemm_wmma_f32<<<grid, 128, 0, s>>>(A, lda, W, ldw, C, ldc, bias, K, Nout, act);
}

extern "C" void kernel_launch(void* const* d_in, const int* in_sizes, int n_in,
                              void* d_out, int out_size, void* d_ws, size_t ws_size,
                              hipStream_t stream) {
    (void)in_sizes; (void)n_in; (void)out_size; (void)ws_size;

    const float* x = (const float*)d_in[0];
    const float* ln_g[3] = {(const float*)d_in[19], (const float*)d_in[21], (const float*)d_in[23]};
    const float* ln_b[3] = {(const float*)d_in[20], (const float*)d_in[22], (const float*)d_in[24]};
    const float* W_ff1 = (const float*)d_in[25];
    const float* b_ff1 = (const float*)d_in[26];
    const float* W_ff2 = (const float*)d_in[27];
    const float* b_ff2 = (const float*)d_in[28];

    // Workspace layout (floats): ~159 MB total
    float* ws   = (float*)d_ws;
    float* xt   = ws;                                  // BL*D
    float* o    = xt   + (size_t)BL * DMODEL;          // BL*D (of + ob accumulator)
    float* bufA = o    + (size_t)BL * DMODEL;          // BL*F  (xz / FFN hidden)
    float* bufB = bufA + (size_t)BL * DFF;             // BL*D  (xc / FFN out)
    float* bufC = bufB + (size_t)BL * DMODEL;          // BL*D  (delta, aliased as y)
    float* bufD = bufC + (size_t)BL * DMODEL;          // BL*D  (mamba GEMM out)
    float* bufP = bufD + (size_t)BL * DMODEL;          // BL*80 (proj)
    float* aP   = bufP + (size_t)BL * PROJW;           // BD*NCHUNK*N
    float* bC   = aP   + (size_t)BD * NCHUNK * DSTATE; // BD*NCHUNK*N
    float* h0   = bC   + (size_t)BD * NCHUNK * DSTATE; // BD*NCHUNK*N

    // 1) (B,D,L) -> (B,L,D)
    transpose_bdl_to_bld<<<dim3(SEQ / 32, DMODEL / 32, BATCH), dim3(32, 8), 0, stream>>>(x, xt);

    // 2) forward (dir=0) and backward (dir=1) Mamba passes
    for (int dir = 0; dir < 2; ++dir) {
        const int base = 1 + dir * 9;
        const float* W_in    = (const float*)d_in[base + 0];
        const float* conv_w  = (const float*)d_in[base + 1];
        const float* conv_b  = (const float*)d_in[base + 2];
        const float* W_xproj = (const float*)d_in[base + 3];
        const float* W_dt    = (const float*)d_in[base + 4];
        const float* b_dt    = (const float*)d_in[base + 5];
        const float* A_log   = (const float*)d_in[base + 6];
        const float* Dskip   = (const float*)d_in[base + 7];
        const float* W_out   = (const float*)d_in[base + 8];

        // xz = xt @ W_in^T
        launch_gemm(xt, DMODEL, W_in, DMODEL, bufA, 2 * DMODEL, nullptr,
                    BL, DMODEL, 2 * DMODEL, 0, stream);
        // xc = silu(conv(xi))
        conv_silu_kernel<<<(BL * DMODEL) / 256, 256, 0, stream>>>(bufA, conv_w, conv_b, bufB, dir);
        // proj = xc @ W_xproj^T
        launch_gemm(bufB, DMODEL, W_xproj, DMODEL, bufP, PROJW, nullptr,
                    BL, DMODEL, PROJW, 0, stream);
        // deltaPre = dt @ W_dt^T + b_dt   (A = proj with row stride 80)
        launch_gemm(bufP, PROJW, W_dt, DTRANK, bufC, DMODEL, b_dt,
                    BL, DTRANK, DMODEL, 0, stream);
        // delta = softplus(deltaPre)
        softplus_kernel<<<(BL * DMODEL) / 256, 256, 0, stream>>>(bufC, BL * DMODEL);
        // chunked SSM scan -> y (aliases delta buffer)
        scan_phase1<<<(BD * NCHUNK) / 256, 256, 0, stream>>>(
            bufC, bufB, bufP, A_log, aP, bC, dir);
        scan_phase2<<<BD / 256, 256, 0, stream>>>(aP, bC, h0);
        scan_phase3<<<(BD * NCHUNK) / 256, 256, 0, stream>>>(
            bufC, bufB, bufP, bufA, A_log, Dskip, h0, bufC, dir);
        // mo = y @ W_out^T
        launch_gemm(bufC, DMODEL, W_out, DMODEL, bufD, DMODEL, nullptr,
                    BL, DMODEL, DMODEL, 0, stream);
        // o = LN(xt + mo)  (dir 0 writes, dir 1 accumulates -> o = of + ob)
        ln_residual_kernel<<<BL, 256, 0, stream>>>(xt, bufD, ln_g[dir], ln_b[dir], o, dir);
    }

    // 3) FFN: h1 = relu(o @ W_ff1^T + b_ff1); f2 = h1 @ W_ff2^T + b_ff2
    launch_gemm(o, DMODEL, W_ff1, DMODEL, bufA, DFF, b_ff1, BL, DMODEL, DFF, 1, stream);
    launch_gemm(bufA, DFF, W_ff2, DFF, bufB, DMODEL, b_ff2, BL, DFF, DMODEL, 0, stream);

    // 4) out = LN(o + f2)
    ln_residual_kernel<<<BL, 256, 0, stream>>>(o, bufB, ln_g[2], ln_b[2], (float*)d_out, 0);
}